// TextPredictor_20882130993220
// MI455X (gfx1250) — compile-verified
//
#include <hip/hip_runtime.h>
#include <math.h>

typedef float v2f __attribute__((ext_vector_type(2)));
typedef float v8f __attribute__((ext_vector_type(8)));

#define B_   8
#define S_   4096
#define D_   1024
#define H_   1024
#define PD_  512
#define P_   4
#define TOK_   2048   // E*L
#define TC_    8      // token chunks
#define TOKPC_ 256    // tokens per chunk

// ---------------------------------------------------------------------------
// Kernel 1: gather rows + partial sum per (token-chunk, batch, dim-chunk).
// grid = (4 dim-chunks, 8 token-chunks, 8 batches), block = 256 threads.
// Deterministic (no float atomics); reads are fully coalesced (256 contiguous
// floats per row slice).
// ---------------------------------------------------------------------------
__global__ void gather_partials(const float* __restrict__ ctx,
                                const int* __restrict__ masks,
                                float* __restrict__ partials) {
  __shared__ int idx[TOKPC_];
  const int dc  = blockIdx.x;   // 0..3
  const int tc  = blockIdx.y;   // 0..7
  const int b   = blockIdx.z;   // 0..7
  const int tid = threadIdx.x;  // 0..255
  idx[tid] = masks[b * TOK_ + tc * TOKPC_ + tid];
  __syncthreads();
  const float* base = ctx + (size_t)b * S_ * D_ + dc * 256 + tid;
  float sum = 0.f;
#pragma unroll 8
  for (int i = 0; i < TOKPC_; ++i) {
    sum += base[(size_t)idx[i] * D_];
  }
  partials[(size_t)(tc * B_ + b) * D_ + dc * 256 + tid] = sum;
}

// ---------------------------------------------------------------------------
// Kernel 2: reduce token-chunk partials -> pooled[16][1024] (rows 8..15 zero
// padding so WMMA M=16 tiles see a clean A matrix).
// ---------------------------------------------------------------------------
__global__ void reduce_pool(const float* __restrict__ partials,
                            float* __restrict__ pooled) {
  const int g   = blockIdx.x * 256 + threadIdx.x;  // 0 .. 16*1024-1
  const int row = g >> 10;
  const int d   = g & 1023;
  float s = 0.f;
  if (row < B_) {
#pragma unroll
    for (int tc = 0; tc < TC_; ++tc)
      s += partials[(size_t)(tc * B_ + row) * D_ + d];
    s *= (1.0f / (float)TOK_);
  }
  pooled[g] = s;
}

// ---------------------------------------------------------------------------
// Kernel 3: GEMM1 + bias + exact GELU via V_WMMA_F32_16X16X4_F32.
// One wave per 16-wide N tile; grid = 64 tiles. A = pooled[16][1024] (padded),
// B = W1[1024][1024], out = h[16][1024].
// A layout (ISA 16x4 f32): lane l -> M = l&15, K = 2*(l>=16) + {0,1}.
// ---------------------------------------------------------------------------
__global__ void gemm1_gelu(const float* __restrict__ A,
                           const float* __restrict__ W1,
                           const float* __restrict__ b1,
                           float* __restrict__ Hout) {
  const int lane = threadIdx.x & 31;
  const int half = lane >> 4;   // 0: K={0,1}, 1: K={2,3}
  const int l16  = lane & 15;
  const int n0   = blockIdx.x * 16;
  const int col  = n0 + l16;

  const float* Ap = A + l16 * D_ + 2 * half;
  const float* Wp = W1 + (size_t)(2 * half) * H_ + col;

  v8f acc = {};
  for (int k = 0; k < D_; k += 4) {
    v2f a, b;
    a.x = Ap[k];
    a.y = Ap[k + 1];
    b.x = Wp[(size_t)k * H_];
    b.y = Wp[(size_t)(k + 1) * H_];
    acc = __builtin_amdgcn_wmma_f32_16x16x4_f32(
        false, a, false, b, (short)0, acc, false, false);
  }

  const float bias = b1[col];
#pragma unroll
  for (int r = 0; r < 8; ++r) {
    float x = acc[r] + bias;
    float g = 0.5f * x * (1.0f + erff(x * 0.70710678118654752f));  // exact GELU
    int row = r + 8 * half;          // D layout: VGPR r -> M = r + 8*half
    Hout[row * H_ + col] = g;
  }
}

// ---------------------------------------------------------------------------
// Kernel 4: per-row LayerNorm over H=1024 (rows 0..7 only; padded rows are
// never consumed by valid output lanes). 256 threads/row, wave32 shfl reduce.
// ---------------------------------------------------------------------------
__global__ void layernorm_rows(float* __restrict__ Hbuf,
                               const float* __restrict__ gam,
                               const float* __restrict__ bet) {
  const int row = blockIdx.x;      // 0..7
  const int tid = threadIdx.x;     // 0..255
  __shared__ float red1[8];
  __shared__ float red2[8];

  float v[4];
  float s = 0.f;
#pragma unroll
  for (int j = 0; j < 4; ++j) {
    v[j] = Hbuf[row * H_ + tid + j * 256];
    s += v[j];
  }
#pragma unroll
  for (int off = 16; off > 0; off >>= 1) s += __shfl_down(s, off, 32);
  if ((tid & 31) == 0) red1[tid >> 5] = s;
  __syncthreads();
  float tot = 0.f;
#pragma unroll
  for (int i = 0; i < 8; ++i) tot += red1[i];
  const float mean = tot * (1.0f / (float)H_);

  float s2 = 0.f;
#pragma unroll
  for (int j = 0; j < 4; ++j) {
    float d = v[j] - mean;
    s2 += d * d;
  }
#pragma unroll
  for (int off = 16; off > 0; off >>= 1) s2 += __shfl_down(s2, off, 32);
  if ((tid & 31) == 0) red2[tid >> 5] = s2;
  __syncthreads();
  float tot2 = 0.f;
#pragma unroll
  for (int i = 0; i < 8; ++i) tot2 += red2[i];
  const float inv = rsqrtf(tot2 * (1.0f / (float)H_) + 1e-5f);

#pragma unroll
  for (int j = 0; j < 4; ++j) {
    int c = tid + j * 256;
    Hbuf[row * H_ + c] = (v[j] - mean) * inv * gam[c] + bet[c];
  }
}

// ---------------------------------------------------------------------------
// Kernel 5: GEMM2 + bias + repeat(P=4) store via V_WMMA_F32_16X16X4_F32.
// grid = 32 tiles of PD=512. Rows 0..7 (batches) live in lanes 0..15.
// ---------------------------------------------------------------------------
__global__ void gemm2_out(const float* __restrict__ Hbuf,
                          const float* __restrict__ W2,
                          const float* __restrict__ b2,
                          float* __restrict__ out) {
  const int lane = threadIdx.x & 31;
  const int half = lane >> 4;
  const int l16  = lane & 15;
  const int n0   = blockIdx.x * 16;
  const int col  = n0 + l16;

  const float* Ap = Hbuf + l16 * H_ + 2 * half;
  const float* Wp = W2 + (size_t)(2 * half) * PD_ + col;

  v8f acc = {};
  for (int k = 0; k < H_; k += 4) {
    v2f a, b;
    a.x = Ap[k];
    a.y = Ap[k + 1];
    b.x = Wp[(size_t)k * PD_];
    b.y = Wp[(size_t)(k + 1) * PD_];
    acc = __builtin_amdgcn_wmma_f32_16x16x4_f32(
        false, a, false, b, (short)0, acc, false, false);
  }

  if (half == 0) {  // lanes 0..15 hold D rows 0..7 == the 8 batch outputs
    const float bias = b2[col];
#pragma unroll
    for (int r = 0; r < 8; ++r) {
      float val = acc[r] + bias;
#pragma unroll
      for (int p = 0; p < P_; ++p)
        out[(size_t)(r * P_ + p) * PD_ + col] = val;
    }
  }
}

// ---------------------------------------------------------------------------
extern "C" void kernel_launch(void* const* d_in, const int* in_sizes, int n_in,
                              void* d_out, int out_size, void* d_ws, size_t ws_size,
                              hipStream_t stream) {
  const float* ctx  = (const float*)d_in[0];  // [8,4096,1024]
  const float* W1   = (const float*)d_in[1];  // [1024,1024]
  const float* b1   = (const float*)d_in[2];  // [1024]
  const float* ln_g = (const float*)d_in[3];  // [1024]
  const float* ln_b = (const float*)d_in[4];  // [1024]
  const float* W2   = (const float*)d_in[5];  // [1024,512]
  const float* b2   = (const float*)d_in[6];  // [512]
  const int*  masks = (const int*)d_in[7];    // [8,4,512]
  (void)in_sizes; (void)n_in; (void)out_size; (void)ws_size;

  float* out = (float*)d_out;                 // [32,512]

  char* ws = (char*)d_ws;
  float* pooled   = (float*)(ws);                      // 16*1024 f32 = 64 KB
  float* partials = (float*)(ws + 65536);              // 8*8*1024 f32 = 256 KB
  float* hbuf     = (float*)(ws + 65536 + 262144);     // 16*1024 f32 = 64 KB

  gather_partials<<<dim3(4, TC_, B_), 256, 0, stream>>>(ctx, masks, partials);
  reduce_pool<<<64, 256, 0, stream>>>(partials, pooled);
  gemm1_gelu<<<H_ / 16, 32, 0, stream>>>(pooled, W1, b1, hbuf);
  layernorm_rows<<<B_, 256, 0, stream>>>(hbuf, ln_g, ln_b);
  gemm2_out<<<PD_ / 16, 32, 0, stream>>>(hbuf, W2, b2, out);
}